// TransformerBlock_47450798686413
// MI455X (gfx1250) — compile-verified
//
#include <hip/hip_runtime.h>
#include <hip/hip_bf16.h>
#include <math.h>

// ---- problem constants (match reference) ----
#define BB   4
#define SS   2048
#define DD   1024
#define HH   16
#define DHH  64
#define DFFF 4096
#define BSS  (BB * SS)   // 8192 tokens

typedef float v2f __attribute__((ext_vector_type(2)));
typedef float v4f __attribute__((ext_vector_type(4)));
typedef float v8f __attribute__((ext_vector_type(8)));

// CDNA5 fp32 matrix op: D(16x16,f32) = A(16x4,f32) x B(4x16,f32) + C
__device__ __forceinline__ v8f wmma4(v2f a, v2f b, v8f c) {
  return __builtin_amdgcn_wmma_f32_16x16x4_f32(
      /*neg_a=*/false, a, /*neg_b=*/false, b,
      /*c_mod=*/(short)0, c, /*reuse_a=*/false, /*reuse_b=*/false);
}

__device__ __forceinline__ float gelu_f(float x) {
  const float c = 0.79788456080286535588f;  // sqrt(2/pi)
  float x3 = x * x * x;
  return 0.5f * x * (1.0f + tanhf(c * (x + 0.044715f * x3)));
}

// ============================================================
// LayerNorm: out = ((x - mean)/sqrt(var + 1e-7) + shift) * scale
// one 256-thread block per row of D=1024
// ============================================================
__global__ void __launch_bounds__(256, 1)
ln_kernel(const float* __restrict__ x,
          const float* __restrict__ scale,
          const float* __restrict__ shift,
          float* __restrict__ out) {
  __shared__ float red[256];
  const int row = blockIdx.x;
  const float* xr = x + (size_t)row * DD;
  float v0[4];
  float s = 0.f;
#pragma unroll
  for (int i = 0; i < 4; ++i) {
    v0[i] = xr[threadIdx.x + i * 256];
    s += v0[i];
  }
  red[threadIdx.x] = s;
  __syncthreads();
  for (int off = 128; off > 0; off >>= 1) {
    if (threadIdx.x < off) red[threadIdx.x] += red[threadIdx.x + off];
    __syncthreads();
  }
  const float mean = red[0] * (1.0f / DD);
  __syncthreads();
  float sq = 0.f;
#pragma unroll
  for (int i = 0; i < 4; ++i) {
    float d = v0[i] - mean;
    sq += d * d;
  }
  red[threadIdx.x] = sq;
  __syncthreads();
  for (int off = 128; off > 0; off >>= 1) {
    if (threadIdx.x < off) red[threadIdx.x] += red[threadIdx.x + off];
    __syncthreads();
  }
  const float var = red[0] * (1.0f / DD);
  const float rstd = rsqrtf(var + 1e-7f);
  float* orow = out + (size_t)row * DD;
#pragma unroll
  for (int i = 0; i < 4; ++i) {
    int d = threadIdx.x + i * 256;
    orow[d] = ((v0[i] - mean) * rstd + shift[d]) * scale[d];
  }
}

// ============================================================
// GEMM: C[M,N] = A[M,K] @ W[N,K]^T  (+bias) (+gelu) (+residual)
// block = 256 thr = 8 waves (4x2); wave tile 32x64; block tile 128x128.
// Loads are batched (12 per 2 k-steps) ahead of the 16 WMMAs so the
// compiler can clause them and hide latency behind the WMMA chain.
// M%128==0, N%128==0, K%8==0 guaranteed by the launch shapes.
// ============================================================
template <bool GELU>
__global__ void __launch_bounds__(256, 1)
gemm_kernel(const float* __restrict__ A,
            const float* __restrict__ W,
            const float* __restrict__ bias,
            const float* __restrict__ resid,
            float* __restrict__ C,
            int M, int N, int K) {
  const int lane = threadIdx.x & 31;
  const int w = threadIdx.x >> 5;          // 0..7
  const int wm = w >> 1, wn = w & 1;       // 4x2 wave grid
  const int row0 = blockIdx.y * 128 + wm * 32;
  const int col0 = blockIdx.x * 128 + wn * 64;
  const int lm = lane & 15;
  const int half = lane >> 4;
  const int koff = half * 2;               // fragment k sub-offset

  const float* a0p = A + (size_t)(row0 + lm) * K + koff;
  const float* a1p = a0p + (size_t)16 * K;
  const float* b0p = W + (size_t)(col0 + lm) * K + koff;
  const float* b1p = b0p + (size_t)16 * K;
  const float* b2p = b0p + (size_t)32 * K;
  const float* b3p = b0p + (size_t)48 * K;

  v8f acc[2][4];
#pragma unroll
  for (int i = 0; i < 2; ++i)
#pragma unroll
    for (int j = 0; j < 4; ++j) acc[i][j] = (v8f){};

  for (int k = 0; k < K; k += 8) {
    v2f af[2][2], bf[2][4];
#pragma unroll
    for (int u = 0; u < 2; ++u) {
      const int kk = k + u * 4;
      af[u][0] = *(const v2f*)(a0p + kk);
      af[u][1] = *(const v2f*)(a1p + kk);
      bf[u][0] = *(const v2f*)(b0p + kk);
      bf[u][1] = *(const v2f*)(b1p + kk);
      bf[u][2] = *(const v2f*)(b2p + kk);
      bf[u][3] = *(const v2f*)(b3p + kk);
    }
#pragma unroll
    for (int u = 0; u < 2; ++u)
#pragma unroll
      for (int i = 0; i < 2; ++i)
#pragma unroll
        for (int j = 0; j < 4; ++j)
          acc[i][j] = wmma4(af[u][i], bf[u][j], acc[i][j]);
  }

  int cc[4];
  float bb[4];
#pragma unroll
  for (int j = 0; j < 4; ++j) {
    cc[j] = col0 + lm + 16 * j;
    bb[j] = bias ? bias[cc[j]] : 0.f;
  }
#pragma unroll
  for (int i = 0; i < 2; ++i) {
#pragma unroll
    for (int v = 0; v < 8; ++v) {
      const int r = row0 + 16 * i + v + 8 * half;
      const size_t rb = (size_t)r * N;
#pragma unroll
      for (int j = 0; j < 4; ++j) {
        float val = acc[i][j][v] + bb[j];
        if (GELU) val = gelu_f(val);
        if (resid) val += resid[rb + cc[j]];
        C[rb + cc[j]] = val;
      }
    }
  }
}

// ============================================================
// Causal flash attention, fp32, DH=64.
// Q/K/V/O laid out as (B*S, D) with head slice at h*DH.
// 4 waves/block, one 16-query tile per wave; all waves share (b,h),
// so each 16x64 K and V tile is cooperatively staged in LDS once per
// block (kt loop is block-uniform; compute predicated by kt <= qt).
// __launch_bounds__(128, 1) lifts the VGPR cap so the preloaded Q
// fragments and ctx accumulators never spill to scratch.
// grid = (S/64, B*H), block = 128
// ============================================================
__global__ void __launch_bounds__(128, 1)
attn_kernel(const float* __restrict__ Q,
            const float* __restrict__ Kc,
            const float* __restrict__ V,
            float* __restrict__ O) {
  __shared__ float kbuf[16][68];     // 16 keys x DH=64, padded stride
  __shared__ float vbuf[16][68];
  __shared__ float pbuf[4][16][17];  // per-wave P tile, padded stride

  const int lane = threadIdx.x & 31;
  const int wid = threadIdx.x >> 5;
  const int qt = blockIdx.x * 4 + wid;     // this wave's query tile
  const int qtmax = blockIdx.x * 4 + 3;    // block-uniform loop bound
  const int bh = blockIdx.y;
  const int b = bh >> 4;                   // / H
  const int h = bh & 15;                   // % H
  const size_t base = (size_t)b * SS * DD + (size_t)h * DHH;
  const float* Qb = Q + base;
  const float* Kb = Kc + base;
  const float* Vb = V + base;
  float* Ob = O + base;

  const int lm = lane & 15;
  const int half = lane >> 4;
  const int koff = half * 2;
  const float qscale = 0.125f;             // 1/sqrt(DH)

  // staging assignment: 128 threads cover 16 rows x 64 floats (8 each)
  const int srow = threadIdx.x >> 3;        // 0..15
  const int scol = (threadIdx.x & 7) * 8;   // 0,8,...,56

  // preload Q fragments for all 16 k-steps over DH=64, pre-scaled
  v2f qf[16];
  {
    const float* qr = Qb + (size_t)(qt * 16 + lm) * DD + koff;
#pragma unroll
    for (int t = 0; t < 16; ++t) {
      v2f tmp = *(const v2f*)(qr + t * 4);
      qf[t].x = tmp.x * qscale;
      qf[t].y = tmp.y * qscale;
    }
  }

  float rmax[8], rsum[8];
#pragma unroll
  for (int v = 0; v < 8; ++v) { rmax[v] = -INFINITY; rsum[v] = 0.f; }
  v8f ctx[4] = {{}, {}, {}, {}};

  for (int kt = 0; kt <= qtmax; ++kt) {
    // ---- cooperative K/V tile staging (block-uniform) ----
    __syncthreads();  // previous iteration's readers are done
    {
      const float* kg = Kb + (size_t)(kt * 16 + srow) * DD + scol;
      const float* vg = Vb + (size_t)(kt * 16 + srow) * DD + scol;
      v4f k0 = *(const v4f*)(kg);
      v4f k1 = *(const v4f*)(kg + 4);
      v4f w0 = *(const v4f*)(vg);
      v4f w1 = *(const v4f*)(vg + 4);
      *(v4f*)(&kbuf[srow][scol])     = k0;
      *(v4f*)(&kbuf[srow][scol + 4]) = k1;
      *(v4f*)(&vbuf[srow][scol])     = w0;
      *(v4f*)(&vbuf[srow][scol + 4]) = w1;
    }
    __syncthreads();

    if (kt <= qt) {
      // ---- S tile = (Q/8) @ K^T : load 16 B-frags, then 16 WMMAs ----
      v2f bkf[16];
#pragma unroll
      for (int t = 0; t < 16; ++t)
        bkf[t] = *(const v2f*)(&kbuf[lm][t * 4 + koff]);
      v8f s = {};
#pragma unroll
      for (int t = 0; t < 16; ++t) s = wmma4(qf[t], bkf[t], s);

      // ---- online softmax (level-parallel shuffle reductions) ----
      float sv[8];
#pragma unroll
      for (int v = 0; v < 8; ++v) {
        sv[v] = s[v];
        if (kt == qt && lm > v + 8 * half) sv[v] = -INFINITY;  // causal
      }
      float mv[8];
#pragma unroll
      for (int v = 0; v < 8; ++v) mv[v] = sv[v];
#pragma unroll
      for (int msk = 1; msk <= 8; msk <<= 1) {
        float tv[8];
#pragma unroll
        for (int v = 0; v < 8; ++v) tv[v] = __shfl_xor(mv[v], msk, 32);
#pragma unroll
        for (int v = 0; v < 8; ++v) mv[v] = fmaxf(mv[v], tv[v]);
      }
      float pv[8], corr[8];
#pragma unroll
      for (int v = 0; v < 8; ++v) {
        float nm = fmaxf(rmax[v], mv[v]);
        corr[v] = __expf(rmax[v] - nm);
        pv[v] = __expf(sv[v] - nm);
        rmax[v] = nm;
      }
      float ts[8];
#pragma unroll
      for (int v = 0; v < 8; ++v) ts[v] = pv[v];
#pragma unroll
      for (int msk = 1; msk <= 8; msk <<= 1) {
        float tv[8];
#pragma unroll
        for (int v = 0; v < 8; ++v) tv[v] = __shfl_xor(ts[v], msk, 32);
#pragma unroll
        for (int v = 0; v < 8; ++v) ts[v] += tv[v];
      }
#pragma unroll
      for (int v = 0; v < 8; ++v) {
        rsum[v] = rsum[v] * corr[v] + ts[v];
        ctx[0][v] *= corr[v];
        ctx[1][v] *= corr[v];
        ctx[2][v] *= corr[v];
        ctx[3][v] *= corr[v];
        pbuf[wid][v + 8 * half][lm] = pv[v];  // stage P (C-layout -> LDS)
      }

      // ---- ctx += P(16x16) @ V(16x64): batch loads, then 16 WMMAs ----
      v2f paf[4];
#pragma unroll
      for (int kk = 0; kk < 4; ++kk) {
        paf[kk].x = pbuf[wid][lm][kk * 4 + koff];
        paf[kk].y = pbuf[wid][lm][kk * 4 + koff + 1];
      }
      v2f bvf[4][4];
#pragma unroll
      for (int kk = 0; kk < 4; ++kk) {
        const int key = kk * 4 + koff;
#pragma unroll
        for (int j = 0; j < 4; ++j) {
          bvf[kk][j].x = vbuf[key][lm + 16 * j];
          bvf[kk][j].y = vbuf[key + 1][lm + 16 * j];
        }
      }
#pragma unroll
      for (int kk = 0; kk < 4; ++kk)
#pragma unroll
        for (int j = 0; j < 4; ++j)
          ctx[j] = wmma4(paf[kk], bvf[kk][j], ctx[j]);
    }
  }

  // ---- normalize and store ctx (16x64) ----
#pragma unroll
  for (int v = 0; v < 8; ++v) {
    const float inv = 1.0f / rsum[v];
    const int r = qt * 16 + v + 8 * half;
    float* orow = Ob + (size_t)r * DD + lm;
    orow[0]  = ctx[0][v] * inv;
    orow[16] = ctx[1][v] * inv;
    orow[32] = ctx[2][v] * inv;
    orow[48] = ctx[3][v] * inv;
  }
}

// ============================================================
// Host launch
// ============================================================
extern "C" void kernel_launch(void* const* d_in, const int* in_sizes, int n_in,
                              void* d_out, int out_size, void* d_ws, size_t ws_size,
                              hipStream_t stream) {
  (void)in_sizes; (void)n_in; (void)out_size; (void)ws_size;
  const float* x    = (const float*)d_in[0];
  const float* Wq   = (const float*)d_in[1];
  const float* Wk   = (const float*)d_in[2];
  const float* Wv   = (const float*)d_in[3];
  const float* Wo   = (const float*)d_in[4];
  const float* bo   = (const float*)d_in[5];
  const float* W1   = (const float*)d_in[6];
  const float* b1   = (const float*)d_in[7];
  const float* W2   = (const float*)d_in[8];
  const float* b2   = (const float*)d_in[9];
  const float* ln1s = (const float*)d_in[10];
  const float* ln1b = (const float*)d_in[11];
  const float* ln2s = (const float*)d_in[12];
  const float* ln2b = (const float*)d_in[13];
  float* out = (float*)d_out;
  float* ws  = (float*)d_ws;

  const size_t NTOK = (size_t)BSS * DD;  // 8M floats
  float* h   = ws;                 // [0, 8M)
  float* q   = ws + NTOK;          // [8M, 16M)
  float* k   = ws + 2 * NTOK;      // [16M, 24M)
  float* v   = ws + 3 * NTOK;      // [24M, 32M)
  float* ctx = h;                  // reuse (h dead after QKV)
  float* x1  = q;                  // reuse (q dead after attention)
  float* h2  = k;                  // reuse (k dead after attention)
  float* ff  = ws + 4 * NTOK;      // [32M, 64M) : BS x DFF

  dim3 blk(256);
  dim3 gD(DD / 128, BSS / 128);     // N=1024 GEMMs
  dim3 gF(DFFF / 128, BSS / 128);   // N=4096 GEMM

  // 1. h = LN1(x)
  ln_kernel<<<BSS, blk, 0, stream>>>(x, ln1s, ln1b, h);
  // 2. q/k/v projections
  gemm_kernel<false><<<gD, blk, 0, stream>>>(h, Wq, nullptr, nullptr, q, BSS, DD, DD);
  gemm_kernel<false><<<gD, blk, 0, stream>>>(h, Wk, nullptr, nullptr, k, BSS, DD, DD);
  gemm_kernel<false><<<gD, blk, 0, stream>>>(h, Wv, nullptr, nullptr, v, BSS, DD, DD);
  // 3. ctx = causal-softmax(q k^T / 8) v   (flash, LDS-shared K/V tiles)
  attn_kernel<<<dim3(SS / 64, BB * HH), dim3(128), 0, stream>>>(q, k, v, ctx);
  // 4. x1 = ctx @ Wo^T + bo + x
  gemm_kernel<false><<<gD, blk, 0, stream>>>(ctx, Wo, bo, x, x1, BSS, DD, DD);
  // 5. h2 = LN2(x1)
  ln_kernel<<<BSS, blk, 0, stream>>>(x1, ln2s, ln2b, h2);
  // 6. ff = gelu(h2 @ W1^T + b1)
  gemm_kernel<true><<<gF, blk, 0, stream>>>(h2, W1, b1, nullptr, ff, BSS, DFFF, DD);
  // 7. out = x1 + ff @ W2^T + b2
  gemm_kernel<false><<<gD, blk, 0, stream>>>(ff, W2, b2, x1, out, BSS, DD, DFFF);
}